// RelationalHypergraphLayer_38603166057018
// MI455X (gfx1250) — compile-verified
//
#include <hip/hip_runtime.h>
#include <hip/hip_bf16.h>
#include <cstdint>

typedef float v2f __attribute__((ext_vector_type(2)));
typedef float v8f __attribute__((ext_vector_type(8)));
typedef unsigned v4u __attribute__((ext_vector_type(4)));
typedef int v4i __attribute__((ext_vector_type(4)));
typedef int v8i __attribute__((ext_vector_type(8)));

#define NN 100000   // N_NODES
#define NH 20000    // N_HYPER
#define NE 1600000  // N_EDGES
#define DIM 128
#define HID 512

// ---------------------------------------------------------------------------
// Degrees: one thread per edge, u32 atomics.
// ---------------------------------------------------------------------------
__global__ void degree_kernel(const int* __restrict__ src, const int* __restrict__ dst,
                              unsigned* __restrict__ degN, unsigned* __restrict__ degH, int E) {
  int e = blockIdx.x * blockDim.x + threadIdx.x;
  if (e < E) {
    atomicAdd(&degN[src[e]], 1u);
    atomicAdd(&degH[dst[e]], 1u);
  }
}

__global__ void rsqrt_deg_kernel(const unsigned* __restrict__ deg, float* __restrict__ out, int n) {
  int i = blockIdx.x * blockDim.x + threadIdx.x;
  if (i < n) {
    unsigned d = deg[i];
    out[i] = rsqrtf((float)(d < 1u ? 1u : d));
  }
}

// ---------------------------------------------------------------------------
// Edge scatter: one wave (32 lanes) per edge. Lane L moves float4 L of the
// 128-float row: coalesced b128 load + 4x global_atomic_add_f32.
// acc[sidx[e]] += rows[gidx[e]] * scale[gidx[e]]
// ---------------------------------------------------------------------------
__global__ __launch_bounds__(256)
void scatter_kernel(const float* __restrict__ rows, const float* __restrict__ scale,
                    const int* __restrict__ gidx, const int* __restrict__ sidx,
                    float* __restrict__ acc, int E) {
  int w = (int)(((unsigned)blockIdx.x * blockDim.x + threadIdx.x) >> 5);
  int lane = threadIdx.x & 31;
  if (w >= E) return;
  int g = gidx[w];
  int s = sidx[w];
  float sc = scale[g];
  const float4 v = ((const float4*)(rows + (size_t)g * DIM))[lane];
  float* p = acc + (size_t)s * DIM + (size_t)lane * 4;
  unsafeAtomicAdd(p + 0, v.x * sc);
  unsafeAtomicAdd(p + 1, v.y * sc);
  unsafeAtomicAdd(p + 2, v.z * sc);
  unsafeAtomicAdd(p + 3, v.w * sc);
}

// ---------------------------------------------------------------------------
// FP32 WMMA GEMM:  C[M,N] = relu?( diag(rowscale) * (A[M,K] @ W[K,N]) + bias )
// (note: (s .* A) @ W == diag(s) (A @ W), so rowscale is applied in the
//  epilogue -- lets the TDM copy raw bytes)
//
// Block = 256 threads = 8 waves. Block tile = 16 rows x 128 cols
// (wave w owns 16x16 tile at col blockIdx.y*128 + w*16).
// A tile (16 x K, K<=512) DMA'd into LDS by the Tensor Data Mover:
// wave 0 builds a 2D D# (tile K x 16, stride K, 4B elements), issues
// tensor_load_to_lds, waits TENSORcnt==0, then the workgroup barrier
// publishes the tile. K-loop steps by 4 using v_wmma_f32_16x16x4_f32.
//
// Fragment layouts per CDNA5 ISA 7.12.2:
//   A 16x4 : lanes 0-15 -> {K=0,K=1}, lanes 16-31 -> {K=2,K=3}, M = lane&15
//   B 4x16 : lanes 0-15 -> {K=0,K=1}, lanes 16-31 -> {K=2,K=3}, N = lane&15
//   C 16x16: VGPR r -> M = r (lanes 0-15) / r+8 (lanes 16-31), N = lane&15
//
// In-place safe when gridDim.y == 1 (A tile fully landed in LDS before any
// C write; no other block touches these 16 rows).
// ---------------------------------------------------------------------------
__global__ __launch_bounds__(256)
void wmma_gemm_kernel(const float* __restrict__ A, const float* __restrict__ W,
                      const float* __restrict__ bias, const float* __restrict__ rowscale,
                      float* __restrict__ C, int K, int N, int M, int relu) {
  __shared__ __align__(16) float As[16 * HID];
  const int m0 = blockIdx.x * 16;
  const int tid = threadIdx.x;

  if (tid < 32) {
    // --- Tensor DMA descriptor (2D): A[m0 .. m0+15, 0 .. K-1] -> LDS ---
    const unsigned lds_addr = (unsigned)(unsigned long long)(uintptr_t)(&As[0]);
    const unsigned long long ga =
        (unsigned long long)(uintptr_t)(A + (size_t)m0 * K);
    v4u g0;
    g0[0] = 1u;                                            // count=1 (valid user D#)
    g0[1] = lds_addr;                                      // lds_addr (bytes)
    g0[2] = (unsigned)(ga & 0xFFFFFFFFull);                // global_addr[31:0]
    g0[3] = (unsigned)((ga >> 32) & 0x01FFFFFFull)         // global_addr[56:32]
            | (2u << 30);                                  // type=2 ("image")
    v8i g1;
    g1[0] = (int)(2u << 16);                               // wg_mask=0, data_size=2 (4B)
    g1[1] = (int)((unsigned)K << 16);                      // tensor_dim0[15:0]=K
    g1[2] = (int)(((unsigned)K >> 16) |                    // tensor_dim0[31:16]
                  ((unsigned)M << 16));                    // tensor_dim1[15:0]
    g1[3] = (int)(((unsigned)M >> 16) |                    // tensor_dim1[31:16]
                  ((unsigned)K << 16));                    // tile_dim0=K
    g1[4] = 16;                                            // tile_dim1=16, tile_dim2=0
    g1[5] = K;                                             // tensor_dim0_stride[31:0]=K
    g1[6] = 0;                                             // stride hi / dim1_stride lo
    g1[7] = 0;
    const v4i gz4 = {0, 0, 0, 0};                          // groups 2/3 unused (2D)
    const v8i gz8 = {0, 0, 0, 0, 0, 0, 0, 0};
    __builtin_amdgcn_tensor_load_to_lds(g0, g1, gz4, gz4, gz8, 0);
    __builtin_amdgcn_s_wait_tensorcnt(0);
  }
  __syncthreads();

  const int wave = tid >> 5;
  const int lane = tid & 31;
  const int col = blockIdx.y * 128 + wave * 16 + (lane & 15);
  const int koff = (lane >> 4) * 2;   // lanes 0-15: K+0/K+1, lanes 16-31: K+2/K+3
  const int row = lane & 15;

  v8f acc = {0.f, 0.f, 0.f, 0.f, 0.f, 0.f, 0.f, 0.f};
  for (int k = 0; k < K; k += 4) {
    v2f a = *(const v2f*)(&As[row * K + k + koff]);        // ds_load_b64
    v2f b;
    b.x = W[(size_t)(k + koff) * N + col];
    b.y = W[(size_t)(k + koff + 1) * N + col];
    acc = __builtin_amdgcn_wmma_f32_16x16x4_f32(
        /*neg_a=*/false, a, /*neg_b=*/false, b,
        /*c_mod=*/(short)0, acc, /*reuse_a=*/false, /*reuse_b=*/false);
  }

  const float bcol = bias ? bias[col] : 0.0f;
  const int rbase = m0 + ((lane >> 4) << 3);
#pragma unroll
  for (int r = 0; r < 8; ++r) {
    const float s = rowscale ? rowscale[rbase + r] : 1.0f;
    float v = acc[r] * s + bcol;
    if (relu) v = fmaxf(v, 0.0f);
    C[(size_t)(rbase + r) * N + col] = v;
  }
}

// ---------------------------------------------------------------------------
// Fused residual + LayerNorm: out[row] = LN(A[row] + B[row]) * gamma + beta.
// One 128-thread block per row; LDS tree reduction. Safe when out aliases B.
// ---------------------------------------------------------------------------
__global__ __launch_bounds__(128)
void add_ln_kernel(const float* __restrict__ A, const float* __restrict__ B,
                   const float* __restrict__ gamma, const float* __restrict__ beta,
                   float* __restrict__ out) {
  const int row = blockIdx.x;
  const int tid = threadIdx.x;
  __shared__ float red[128];
  const size_t idx = (size_t)row * DIM + tid;
  const float x = A[idx] + B[idx];

  red[tid] = x;
  __syncthreads();
  for (int s = 64; s > 0; s >>= 1) {
    if (tid < s) red[tid] += red[tid + s];
    __syncthreads();
  }
  const float mu = red[0] * (1.0f / DIM);
  __syncthreads();

  const float d = x - mu;
  red[tid] = d * d;
  __syncthreads();
  for (int s = 64; s > 0; s >>= 1) {
    if (tid < s) red[tid] += red[tid + s];
    __syncthreads();
  }
  const float var = red[0] * (1.0f / DIM);
  out[idx] = d * rsqrtf(var + 1e-5f) * gamma[tid] + beta[tid];
}

// ---------------------------------------------------------------------------
// Host launch
// ---------------------------------------------------------------------------
extern "C" void kernel_launch(void* const* d_in, const int* in_sizes, int n_in,
                              void* d_out, int out_size, void* d_ws, size_t ws_size,
                              hipStream_t stream) {
  const float* h   = (const float*)d_in[0];
  const int*  src  = (const int*)d_in[1];
  const int*  dst  = (const int*)d_in[2];
  const float* W1  = (const float*)d_in[3];
  const float* b1  = (const float*)d_in[4];
  const float* W2  = (const float*)d_in[5];
  const float* b2  = (const float*)d_in[6];
  const float* W3  = (const float*)d_in[7];
  const float* b3  = (const float*)d_in[8];
  const float* W4  = (const float*)d_in[9];
  const float* b4  = (const float*)d_in[10];
  const float* g1  = (const float*)d_in[11];
  const float* be1 = (const float*)d_in[12];
  const float* g2  = (const float*)d_in[13];
  const float* be2 = (const float*)d_in[14];
  float* out = (float*)d_out;

  // Workspace carve-up (floats), ~319 MB total.
  float* accH = (float*)d_ws;                 // [NH, DIM]  -> hHyper (in place)
  float* accN = accH + (size_t)NH * DIM;      // [NN, DIM]  -> hNode  (in place)
  float* h1   = accN + (size_t)NN * DIM;      // [NN, DIM]
  float* t    = h1   + (size_t)NN * DIM;      // [NN, HID]
  float* cn   = t    + (size_t)NN * HID;      // [NN]
  float* ch   = cn   + NN;                    // [NH]
  unsigned* degN = (unsigned*)(ch + NH);      // [NN]
  unsigned* degH = degN + NN;                 // [NH]

  // Zero accumulators + degree counters (contiguous blocks).
  (void)hipMemsetAsync(accH, 0, (size_t)(NH + NN) * DIM * sizeof(float), stream);
  (void)hipMemsetAsync(degN, 0, (size_t)(NN + NH) * sizeof(unsigned), stream);

  // 1) degrees + symmetric norms
  degree_kernel<<<(NE + 255) / 256, 256, 0, stream>>>(src, dst, degN, degH, NE);
  rsqrt_deg_kernel<<<(NN + 255) / 256, 256, 0, stream>>>(degN, cn, NN);
  rsqrt_deg_kernel<<<(NH + 255) / 256, 256, 0, stream>>>(degH, ch, NH);

  // 2) node -> hyperedge aggregate:  accH[dst] += h[src] * cn[src]
  scatter_kernel<<<(NE + 7) / 8, 256, 0, stream>>>(h, cn, src, dst, accH, NE);

  // 3) hHyper = (accH * ch) @ W1 + b1   (in place on accH)
  wmma_gemm_kernel<<<dim3(NH / 16, 1), 256, 0, stream>>>(accH, W1, b1, ch, accH, DIM, DIM, NH, 0);

  // 4) hyperedge -> node aggregate:  accN[src] += hHyper[dst] * ch[dst]
  scatter_kernel<<<(NE + 7) / 8, 256, 0, stream>>>(accH, ch, dst, src, accN, NE);

  // 5) hNode = (accN * cn) @ W2 + b2   (in place on accN)
  wmma_gemm_kernel<<<dim3(NN / 16, 1), 256, 0, stream>>>(accN, W2, b2, cn, accN, DIM, DIM, NN, 0);

  // 6) h1 = LN(h + hNode)
  add_ln_kernel<<<NN, 128, 0, stream>>>(h, accN, g1, be1, h1);

  // 7) t = relu(h1 @ W3 + b3)   [NN, HID]
  wmma_gemm_kernel<<<dim3(NN / 16, HID / 128), 256, 0, stream>>>(h1, W3, b3, nullptr, t, DIM, HID, NN, 1);

  // 8) f = t @ W4 + b4  -> stored in d_out
  wmma_gemm_kernel<<<dim3(NN / 16, 1), 256, 0, stream>>>(t, W4, b4, nullptr, out, HID, DIM, NN, 0);

  // 9) out = LN(h1 + f)   (reads/writes d_out, block-local safe)
  add_ln_kernel<<<NN, 128, 0, stream>>>(h1, out, g2, be2, out);
}